// EfficientGATConv_85048942396182
// MI455X (gfx1250) — compile-verified
//
#include <hip/hip_runtime.h>
#include <hip/hip_bf16.h>
#include <math.h>

typedef __attribute__((ext_vector_type(2))) float v2f;
typedef __attribute__((ext_vector_type(8))) float v8f;

#define IN_F 64
#define HD   128   // H*D
#define H_   8
#define D_   16

// ---------------------------------------------------------------------------
// Kernel 1: projection GEMM via f32 WMMA.
//   blockIdx.y == 0 : ft  = feat @ W     -> ft_ws
//   blockIdx.y == 1 : res = feat @ Wres  -> d_out (residual pre-seeded)
// One wave per 16x16 output tile; 8 waves cover the 128 output columns.
// ---------------------------------------------------------------------------
__global__ __launch_bounds__(256) void gat_proj_wmma(
    const float* __restrict__ feat,
    const float* __restrict__ W,
    const float* __restrict__ Wres,
    float* __restrict__ ft_ws,
    float* __restrict__ res_out,
    int nrows)
{
    const int wave = threadIdx.x >> 5;   // 0..7 -> column tile (== head)
    const int lane = threadIdx.x & 31;
    const int row0 = blockIdx.x * 16;
    const int col0 = wave * 16;

    const float* B  = (blockIdx.y == 0) ? W : Wres;
    float* outp     = (blockIdx.y == 0) ? ft_ws : res_out;

    const int r  = lane & 15;
    const int hi = lane >> 4;            // 0: K pair {0,1}; 1: K pair {2,3}

    // Clamp load row so EXEC stays all-ones through the WMMA (tail safety).
    const int arow = (row0 + r < nrows) ? (row0 + r) : (nrows - 1);
    const float* apBase = feat + (size_t)arow * IN_F;

    v8f acc = {};
#pragma unroll
    for (int kb = 0; kb < IN_F; kb += 4) {
        // A tile 16x4 (f32): lanes 0-15 hold K={kb,kb+1}, lanes 16-31 K={kb+2,kb+3}
        v2f a;
        a.x = apBase[kb + 2 * hi + 0];
        a.y = apBase[kb + 2 * hi + 1];
        // B tile 4x16: same K striping, N = col0 + r, row-major W [64 x 128]
        const float* bp = B + (size_t)(kb + 2 * hi) * HD + col0 + r;
        v2f b;
        b.x = bp[0];
        b.y = bp[HD];
        acc = __builtin_amdgcn_wmma_f32_16x16x4_f32(
            /*neg_a=*/false, a, /*neg_b=*/false, b,
            /*c_mod=*/(short)0, acc, /*reuse_a=*/false, /*reuse_b=*/false);
    }

    // C/D layout: VGPR v -> row (v + 8*hi), col = col0 + r
    const int rbase = row0 + 8 * hi;
#pragma unroll
    for (int v = 0; v < 8; ++v) {
        int row = rbase + v;
        if (row < nrows)
            outp[(size_t)row * HD + col0 + r] = acc[v];
    }
}

// ---------------------------------------------------------------------------
// Kernel 2: el/er per (node, head); init emax/denom.
// ---------------------------------------------------------------------------
__global__ void gat_node_attn(
    const float* __restrict__ ft,
    const float* __restrict__ attn_l,
    const float* __restrict__ attn_r,
    float* __restrict__ el, float* __restrict__ er,
    float* __restrict__ emax, float* __restrict__ denom,
    int totalNH)
{
    int i = blockIdx.x * blockDim.x + threadIdx.x;
    if (i >= totalNH) return;
    int h = i & (H_ - 1);
    const float* f  = ft + (size_t)(i >> 3) * HD + h * D_;
    const float* al = attn_l + h * D_;
    const float* ar = attn_r + h * D_;
    float sl = 0.f, sr = 0.f;
#pragma unroll
    for (int d = 0; d < D_; ++d) {
        float v = f[d];
        sl = fmaf(v, al[d], sl);
        sr = fmaf(v, ar[d], sr);
    }
    el[i] = sl;
    er[i] = sr;
    emax[i]  = -INFINITY;
    denom[i] = 0.f;
}

__device__ __forceinline__ float lrelu(float x) {
    return x > 0.f ? x : 0.01f * x;
}

// ---------------------------------------------------------------------------
// Kernel 3: segment max over dst via float atomicMax (global_atomic_max_num_f32)
// one thread per (edge, head)
// ---------------------------------------------------------------------------
__global__ void gat_edge_max(
    const int* __restrict__ src, const int* __restrict__ dst,
    const float* __restrict__ el, const float* __restrict__ er,
    float* __restrict__ emax, int totalEH)
{
    int i = blockIdx.x * blockDim.x + threadIdx.x;
    if (i >= totalEH) return;
    int e = i >> 3, h = i & (H_ - 1);
    int s = src[e], d = dst[e];
    float x = lrelu(el[s * H_ + h] + er[d * H_ + h]);
    atomicMax(&emax[d * H_ + h], x);
}

// ---------------------------------------------------------------------------
// Kernel 4: softmax denominator
// ---------------------------------------------------------------------------
__global__ void gat_edge_sum(
    const int* __restrict__ src, const int* __restrict__ dst,
    const float* __restrict__ el, const float* __restrict__ er,
    const float* __restrict__ emax, float* __restrict__ denom, int totalEH)
{
    int i = blockIdx.x * blockDim.x + threadIdx.x;
    if (i >= totalEH) return;
    int e = i >> 3, h = i & (H_ - 1);
    int s = src[e], d = dst[e];
    float x = lrelu(el[s * H_ + h] + er[d * H_ + h]);
    float ex = expf(x - emax[d * H_ + h]);
    atomicAdd(&denom[d * H_ + h], ex);
}

// ---------------------------------------------------------------------------
// Kernel 5: weighted scatter-sum aggregation (on top of residual in d_out)
// one thread per (edge, head); ft/out rows live in the 192MB L2.
// ---------------------------------------------------------------------------
__global__ void gat_edge_aggr(
    const int* __restrict__ src, const int* __restrict__ dst,
    const float* __restrict__ el, const float* __restrict__ er,
    const float* __restrict__ emax, const float* __restrict__ denom,
    const float* __restrict__ ft, float* __restrict__ out, int totalEH)
{
    int i = blockIdx.x * blockDim.x + threadIdx.x;
    if (i >= totalEH) return;
    int e = i >> 3, h = i & (H_ - 1);
    int s = src[e], d = dst[e];
    float x = lrelu(el[s * H_ + h] + er[d * H_ + h]);
    float a = expf(x - emax[d * H_ + h]) / denom[d * H_ + h];
    const float* fs = ft  + (size_t)s * HD + h * D_;
    float*       od = out + (size_t)d * HD + h * D_;
#pragma unroll
    for (int dd = 0; dd < D_; ++dd)
        atomicAdd(&od[dd], fs[dd] * a);
}

// ---------------------------------------------------------------------------
// Kernel 6: ELU in place
// ---------------------------------------------------------------------------
__global__ void gat_elu(float* __restrict__ out, int total)
{
    int i = blockIdx.x * blockDim.x + threadIdx.x;
    if (i >= total) return;
    float x = out[i];
    out[i] = x > 0.f ? x : expm1f(x);
}

// ---------------------------------------------------------------------------
extern "C" void kernel_launch(void* const* d_in, const int* in_sizes, int n_in,
                              void* d_out, int out_size, void* d_ws, size_t ws_size,
                              hipStream_t stream)
{
    const float* feat   = (const float*)d_in[0];
    const int*   src    = (const int*)  d_in[1];
    const int*   dst    = (const int*)  d_in[2];
    const float* W      = (const float*)d_in[3];
    const float* attn_l = (const float*)d_in[4];
    const float* attn_r = (const float*)d_in[5];
    const float* Wres   = (const float*)d_in[6];
    float* out = (float*)d_out;

    const int N = in_sizes[0] / IN_F;
    const int E = in_sizes[1];

    // workspace layout
    char* ws = (char*)d_ws;
    float* ft    = (float*)ws;                       ws += (size_t)N * HD * sizeof(float);
    float* el    = (float*)ws;                       ws += (size_t)N * H_ * sizeof(float);
    float* er    = (float*)ws;                       ws += (size_t)N * H_ * sizeof(float);
    float* emax  = (float*)ws;                       ws += (size_t)N * H_ * sizeof(float);
    float* denom = (float*)ws;

    // 1) projections (ft -> ws, residual -> d_out)
    dim3 gproj((N + 15) / 16, 2);
    gat_proj_wmma<<<gproj, 256, 0, stream>>>(feat, W, Wres, ft, out, N);

    // 2) el/er + init
    int totalNH = N * H_;
    gat_node_attn<<<(totalNH + 255) / 256, 256, 0, stream>>>(
        ft, attn_l, attn_r, el, er, emax, denom, totalNH);

    // 3-5) edge phase
    int totalEH = E * H_;
    int gE = (totalEH + 255) / 256;
    gat_edge_max <<<gE, 256, 0, stream>>>(src, dst, el, er, emax, totalEH);
    gat_edge_sum <<<gE, 256, 0, stream>>>(src, dst, el, er, emax, denom, totalEH);
    gat_edge_aggr<<<gE, 256, 0, stream>>>(src, dst, el, er, emax, denom, ft, out, totalEH);

    // 6) ELU
    int totalOut = N * HD;
    gat_elu<<<(totalOut + 255) / 256, 256, 0, stream>>>(out, totalOut);
}